// AtomAttentionEncoder_91027536871994
// MI455X (gfx1250) — compile-verified
//
#include <hip/hip_runtime.h>
#include <math.h>

// ---------------------------------------------------------------------------
// AtomAttentionEncoder forward for MI455X (gfx1250, wave32).
// Memory-bound on plm (2048x2048x16 f32 = 256MB) -> fp32 everywhere (exact
// vs reference, and precision is free when HBM-bound at 23.3 TB/s).
// Matrix work on V_WMMA_F32_16X16X4_F32.
// ---------------------------------------------------------------------------

#define N_ATOM  2048
#define N_TOKEN 256
#define C_ATOM  128
#define C_PAIR  16
#define C_TOKEN 384
#define NHEADS  4
#define DHEAD   32

typedef float v2f __attribute__((ext_vector_type(2)));
typedef float v8f __attribute__((ext_vector_type(8)));

// ---------------------------------------------------------------------------
// Generic fp32 GEMM: C[M,N] = A[M,K] @ W[K,N] (+ bias[N]).
// One wave computes one 16x16 tile via V_WMMA_F32_16X16X4_F32 over K/4 steps.
// Fragment layouts per CDNA5 ISA 7.12.2:
//   A 16x4 : lanes 0-15 = rows M, VGPR0/1 hold K={0,1} (lanes<16) or K={2,3}
//   B 4x16 : VGPR v holds row K=v (lanes 0-15) / K=v+2 (lanes 16-31)
//   D 16x16: VGPR r -> M=r (lanes 0-15) / M=r+8 (lanes 16-31), N=lane&15
// Inner loop uses pointer increments only (3x 64-bit add per step) so the
// VALU doesn't drown the WMMA issue slots in address math.
// M multiple of 64, N of 16, K of 4 (true for all call sites).
// ---------------------------------------------------------------------------
__global__ void gemm_wmma(const float* __restrict__ A, const float* __restrict__ W,
                          const float* __restrict__ bias, float* __restrict__ C,
                          int M, int K, int N) {
  int wave = threadIdx.x >> 5;
  int lane = threadIdx.x & 31;
  int m0 = (blockIdx.y * 4 + wave) * 16;
  int n0 = blockIdx.x * 16;
  if (m0 >= M) return;                    // wave-uniform: EXEC stays all-ones
  int row = m0 + (lane & 15);
  int col = n0 + (lane & 15);
  int kh  = (lane < 16) ? 0 : 2;
  const float* ap  = A + (size_t)row * K + kh;      // a.x/a.y contiguous -> b64
  const float* wp0 = W + (size_t)kh * N + col;
  const float* wp1 = wp0 + N;
  const size_t wstep = (size_t)4 * N;
  const size_t wpre  = (size_t)16 * N;              // prefetch 4 steps ahead
  v8f acc = {};
  #pragma unroll 2
  for (int k = 0; k < K; k += 4) {
    v2f a, b;
    a.x = ap[0];
    a.y = ap[1];
    b.x = *wp0;
    b.y = *wp1;
    __builtin_prefetch(wp0 + wpre, 0, 1);           // global_prefetch_b8
    acc = __builtin_amdgcn_wmma_f32_16x16x4_f32(false, a, false, b,
                                                (short)0, acc, false, false);
    ap += 4; wp0 += wstep; wp1 += wstep;
  }
  int rb = (lane < 16) ? 0 : 8;
  float bc = bias ? bias[col] : 0.0f;
  float* cp = C + (size_t)(m0 + rb) * N + col;
  for (int r = 0; r < 8; ++r) { *cp = acc[r] + bc; cp += N; }
}

// ---------------------------------------------------------------------------
// LayerNorm over last dim: one wave per row, shuffle tree reductions
// (wave32-native). g/b may be null (ln0). blockDim = 128 -> 4 rows/block.
// ---------------------------------------------------------------------------
__global__ void k_ln(const float* __restrict__ x, const float* __restrict__ g,
                     const float* __restrict__ b, float* __restrict__ y,
                     int M, int C) {
  int wave = threadIdx.x >> 5, lane = threadIdx.x & 31;
  int r = blockIdx.x * 4 + wave;
  if (r >= M) return;
  const float* xr = x + (size_t)r * C;
  float s = 0.f;
  for (int c = lane; c < C; c += 32) s += xr[c];
  for (int o = 16; o > 0; o >>= 1) s += __shfl_xor(s, o, 32);
  float m = s / (float)C;
  float v = 0.f;
  for (int c = lane; c < C; c += 32) { float d = xr[c] - m; v += d * d; }
  for (int o = 16; o > 0; o >>= 1) v += __shfl_xor(v, o, 32);
  float rs = rsqrtf(v / (float)C + 1e-5f);
  float* yr = y + (size_t)r * C;
  for (int c = lane; c < C; c += 32) {
    float n = (xr[c] - m) * rs;
    yr[c] = g ? (n * g[c] + b[c]) : n;
  }
}

// an = sigmoid(gp + bg[c]) * ln0a + sp
__global__ void k_adaln(const float* __restrict__ gp, const float* __restrict__ bg,
                        const float* __restrict__ ln0a, const float* __restrict__ sp,
                        float* __restrict__ out, int total, int C) {
  int i = blockIdx.x * blockDim.x + threadIdx.x;
  if (i >= total) return;
  int c = i % C;
  float s = 1.f / (1.f + expf(-(gp[i] + bg[c])));
  out[i] = s * ln0a[i] + sp[i];
}

__global__ void k_sigmoid(float* __restrict__ x, int n) {
  int i = blockIdx.x * blockDim.x + threadIdx.x;
  if (i < n) x[i] = 1.f / (1.f + expf(-x[i]));
}
__global__ void k_mul(const float* __restrict__ a, const float* __restrict__ b,
                      float* __restrict__ o, int n) {
  int i = blockIdx.x * blockDim.x + threadIdx.x;
  if (i < n) o[i] = a[i] * b[i];
}
__global__ void k_silu_mul(const float* __restrict__ a, const float* __restrict__ b,
                           float* __restrict__ o, int n) {
  int i = blockIdx.x * blockDim.x + threadIdx.x;
  if (i < n) { float s = a[i] / (1.f + expf(-a[i])); o[i] = s * b[i]; }
}
__global__ void k_fma3(const float* __restrict__ a, const float* __restrict__ b,
                       const float* __restrict__ c, float* __restrict__ o, int n) {
  int i = blockIdx.x * blockDim.x + threadIdx.x;
  if (i < n) o[i] = a[i] + b[i] * c[i];
}
__global__ void k_relu(const float* __restrict__ a, float* __restrict__ o, int n) {
  int i = blockIdx.x * blockDim.x + threadIdx.x;
  if (i < n) o[i] = fmaxf(a[i], 0.f);
}

// ---------------------------------------------------------------------------
// Atom feature embedding: one block per atom, thread per output channel.
// cl0 = feats@W_feats (concat done implicitly; scalar per-atom features are
// lane-invariant -> SGPR broadcast loads).
// ql = cl0 + rl@W_r ; cl = cl0 + s_tok[tok] ; a0 = ql.
// ---------------------------------------------------------------------------
__global__ void k_embed(const float* __restrict__ pos, const float* __restrict__ mask,
                        const float* __restrict__ elem, const float* __restrict__ charge,
                        const float* __restrict__ chars, const float* __restrict__ uidf,
                        const float* __restrict__ rlv, const float* __restrict__ Wf,
                        const float* __restrict__ Wr, const float* __restrict__ s_tok,
                        const int* __restrict__ tok,
                        float* __restrict__ out_ql, float* __restrict__ out_cl,
                        float* __restrict__ a0) {
  int atom = blockIdx.x;
  int c = threadIdx.x;                    // 128 channels
  float s = 0.f;
  s += pos[atom * 3 + 0] * Wf[0 * C_ATOM + c];
  s += pos[atom * 3 + 1] * Wf[1 * C_ATOM + c];
  s += pos[atom * 3 + 2] * Wf[2 * C_ATOM + c];
  s += mask[atom] * Wf[3 * C_ATOM + c];
  for (int e = 0; e < 128; ++e)
    s += elem[(size_t)atom * 128 + e] * Wf[(size_t)(4 + e) * C_ATOM + c];
  s += charge[atom] * Wf[132 * C_ATOM + c];
  for (int t = 0; t < 256; ++t)
    s += chars[(size_t)atom * 256 + t] * Wf[(size_t)(133 + t) * C_ATOM + c];
  s += uidf[atom] * Wf[389 * C_ATOM + c];
  float ql = s + rlv[atom * 3 + 0] * Wr[0 * C_ATOM + c]
               + rlv[atom * 3 + 1] * Wr[1 * C_ATOM + c]
               + rlv[atom * 3 + 2] * Wr[2 * C_ATOM + c];
  float cl = s + s_tok[(size_t)tok[atom] * C_ATOM + c];
  size_t o = (size_t)atom * C_ATOM + c;
  out_ql[o] = ql; a0[o] = ql; out_cl[o] = cl;
}

// z[i,j,:] = LN(zij[i,j,:]) @ W_z   (thread per (i,j), W_z staged in LDS)
__global__ void k_zproj(const float* __restrict__ zij, const float* __restrict__ lnzg,
                        const float* __restrict__ lnzb, const float* __restrict__ Wz,
                        float* __restrict__ z) {
  __shared__ float wz[256];
  int i = blockIdx.x, j = threadIdx.x;
  wz[j] = Wz[j];
  __syncthreads();
  const float* x = zij + ((size_t)i * N_TOKEN + j) * C_PAIR;
  float m = 0.f;
  float xv[16];
  for (int c = 0; c < 16; ++c) { xv[c] = x[c]; m += xv[c]; }
  m *= (1.f / 16.f);
  float v = 0.f;
  for (int c = 0; c < 16; ++c) { float d = xv[c] - m; v += d * d; }
  v *= (1.f / 16.f);
  float rs = rsqrtf(v + 1e-5f);
  float nv[16];
  for (int c = 0; c < 16; ++c) nv[c] = (xv[c] - m) * rs * lnzg[c] + lnzb[c];
  float* o = z + ((size_t)i * N_TOKEN + j) * C_PAIR;
  for (int oc = 0; oc < 16; ++oc) {
    float acc = 0.f;
    for (int c = 0; c < 16; ++c) acc += nv[c] * wz[c * 16 + oc];
    o[oc] = acc;
  }
}

// ---------------------------------------------------------------------------
// Fused pair tensor: base(geometry + z + cwl/cwm) then 3-layer 16x16 relu-MLP
// residual, via WMMA f32 16x16x4. 8 waves/block; each wave owns a tile of
// 16 (i, j..j+15) pairs staged in LDS (pairs = M rows, channels = K/N).
// ---------------------------------------------------------------------------
#define NP 17
__device__ inline v8f mlp_layer(const float (*src)[NP], const float* w, int lane) {
  int rrow = lane & 15;
  int kh   = (lane < 16) ? 0 : 2;
  int ncol = lane & 15;
  v8f acc = {};
  for (int kk = 0; kk < 16; kk += 4) {
    v2f a, b;
    a.x = fmaxf(src[rrow][kk + kh], 0.f);       // relu applied on A-load
    a.y = fmaxf(src[rrow][kk + kh + 1], 0.f);
    b.x = w[(kk + kh) * 16 + ncol];
    b.y = w[(kk + kh + 1) * 16 + ncol];
    acc = __builtin_amdgcn_wmma_f32_16x16x4_f32(false, a, false, b,
                                                (short)0, acc, false, false);
  }
  return acc;
}

__global__ void k_plm(const float* __restrict__ pos, const float* __restrict__ uidf,
                      const int* __restrict__ tok,
                      const float* __restrict__ Woff, const float* __restrict__ Winv,
                      const float* __restrict__ Wvm, const float* __restrict__ z,
                      const float* __restrict__ cwl, const float* __restrict__ cwm,
                      const float* __restrict__ W1, const float* __restrict__ W2,
                      const float* __restrict__ W3, float* __restrict__ plm) {
  __shared__ float xb[8][16][NP];   // base values (pre-relu), per wave
  __shared__ float hb[8][16][NP];   // hidden, per wave
  __shared__ float w1[256], w2[256], w3[256];
  int tid = threadIdx.x;
  w1[tid] = W1[tid]; w2[tid] = W2[tid]; w3[tid] = W3[tid];
  __syncthreads();
  int wv = tid >> 5, lane = tid & 31;
  int i  = blockIdx.y;
  int p  = lane & 15;
  int j  = blockIdx.x * 128 + wv * 16 + p;
  int ch0 = (lane < 16) ? 0 : 8;
  // base features for pair (i, j), channels ch0..ch0+7
  float dx = pos[i * 3 + 0] - pos[j * 3 + 0];
  float dy = pos[i * 3 + 1] - pos[j * 3 + 1];
  float dz = pos[i * 3 + 2] - pos[j * 3 + 2];
  float vm = (uidf[i] == uidf[j]) ? 1.f : 0.f;
  float inv = 1.f / (1.f + sqrtf(dx * dx + dy * dy + dz * dz));
  size_t zb = ((size_t)tok[i] * N_TOKEN + tok[j]) * C_PAIR;
  for (int c = 0; c < 8; ++c) {
    int ch = ch0 + c;
    float bse = vm * (dx * Woff[ch] + dy * Woff[16 + ch] + dz * Woff[32 + ch]
                      + inv * Winv[ch] + Wvm[ch]);
    bse += z[zb + ch] + cwl[(size_t)i * 16 + ch] + cwm[(size_t)j * 16 + ch];
    xb[wv][p][ch] = bse;
  }
  __syncthreads();
  int ncol = lane & 15;
  int rb   = (lane < 16) ? 0 : 8;
  v8f h = mlp_layer(xb[wv], w1, lane);                     // h1
  for (int r = 0; r < 8; ++r) hb[wv][rb + r][ncol] = h[r];
  __syncthreads();
  h = mlp_layer(hb[wv], w2, lane);                         // h2
  __syncthreads();
  for (int r = 0; r < 8; ++r) hb[wv][rb + r][ncol] = h[r];
  __syncthreads();
  h = mlp_layer(hb[wv], w3, lane);                         // h3
  // plm = base + h3   (D frag: row = pair rb+r, col = channel ncol)
  int jbase = blockIdx.x * 128 + wv * 16;
  float* op = plm + ((size_t)i * N_ATOM + (size_t)(jbase + rb)) * C_PAIR + ncol;
  for (int r = 0; r < 8; ++r) {
    *op = xb[wv][rb + r][ncol] + h[r];
    op += C_PAIR;
  }
}

// ---------------------------------------------------------------------------
// Local attention: block per (32-query tile, head pair). The blm mask reduces
// softmax to exactly the 128-key window [32*floor(q/32)-48, +80); masked
// logits underflow to 0 in exp so computing only the window is exact.
// Logits (qk/sqrt(dh) + LN(plm)@Wb) staged in LDS, softmax+PV per row.
// ---------------------------------------------------------------------------
__global__ void k_attn(const float* __restrict__ q, const float* __restrict__ kx,
                       const float* __restrict__ vx, const float* __restrict__ g,
                       const float* __restrict__ plm, const float* __restrict__ lnpg,
                       const float* __restrict__ lnpb, const float* __restrict__ Wb,
                       float* __restrict__ og) {
  __shared__ float lg[2][32][128];
  __shared__ float Gm[16][4];
  __shared__ float Bh[4];
  int t = threadIdx.x;                    // 64 threads
  { int c = t >> 2, h = t & 3; Gm[c][h] = lnpg[c] * Wb[c * 4 + h]; }
  if (t < 4) {
    float s = 0.f;
    for (int c = 0; c < 16; ++c) s += lnpb[c] * Wb[c * 4 + t];
    Bh[t] = s;
  }
  __syncthreads();
  int hp = blockIdx.y * 2;                // head pair
  int q0 = blockIdx.x * 32;
  int k0 = q0 - 48;                       // window = exactly the blm-allowed keys
  const float scale = 0.17677669529663687f;   // 1/sqrt(32)
  // stage 1: logits for heads hp, hp+1
  for (int e = t; e < 32 * 128; e += 64) {
    int qq = e >> 7, kk = e & 127;
    int qa = q0 + qq, ka = k0 + kk;
    float l0 = -3.0e38f, l1 = -3.0e38f;
    if (ka >= 0 && ka < N_ATOM) {
      const float* pr = plm + ((size_t)qa * N_ATOM + ka) * C_PAIR;
      float xv[16]; float m = 0.f;
      for (int c = 0; c < 16; ++c) { xv[c] = pr[c]; m += xv[c]; }
      m *= (1.f / 16.f);
      float var = 0.f;
      for (int c = 0; c < 16; ++c) { float d = xv[c] - m; var += d * d; }
      var *= (1.f / 16.f);
      float rs = rsqrtf(var + 1e-5f);
      float p0 = Bh[hp], p1 = Bh[hp + 1];
      for (int c = 0; c < 16; ++c) {
        float n = (xv[c] - m) * rs;
        p0 += n * Gm[c][hp]; p1 += n * Gm[c][hp + 1];
      }
      const float* qr = q  + (size_t)qa * C_ATOM + hp * DHEAD;
      const float* kr = kx + (size_t)ka * C_ATOM + hp * DHEAD;
      float d0 = 0.f, d1 = 0.f;
      for (int d = 0; d < DHEAD; ++d) {
        d0 += qr[d] * kr[d];
        d1 += qr[DHEAD + d] * kr[DHEAD + d];
      }
      l0 = d0 * scale + p0; l1 = d1 * scale + p1;
    }
    lg[0][qq][kk] = l0; lg[1][qq][kk] = l1;
  }
  __syncthreads();
  // stage 2: softmax + PV, one (q, head) row per thread
  {
    int hl = t >> 5;                      // 0/1 within head-pair
    int hh = hp + hl;
    int qq = t & 31;
    int qa = q0 + qq;
    float mx = -3.0e38f;
    for (int kk = 0; kk < 128; ++kk) mx = fmaxf(mx, lg[hl][qq][kk]);
    float acc[DHEAD];
    for (int d = 0; d < DHEAD; ++d) acc[d] = 0.f;
    float sum = 0.f;
    for (int kk = 0; kk < 128; ++kk) {
      int ka = k0 + kk;
      if (ka < 0 || ka >= N_ATOM) continue;
      float w = expf(lg[hl][qq][kk] - mx);
      sum += w;
      const float* vr = vx + (size_t)ka * C_ATOM + hh * DHEAD;
      for (int d = 0; d < DHEAD; ++d) acc[d] += w * vr[d];
    }
    float is = 1.f / sum;
    const float* gr = g + (size_t)qa * C_ATOM + hh * DHEAD;
    float* orow = og + (size_t)qa * C_ATOM + hh * DHEAD;
    for (int d = 0; d < DHEAD; ++d) orow[d] = gr[d] * acc[d] * is;
  }
}

// ---------------------------------------------------------------------------
// atom -> token segment mean of relu(x). Deterministic (no float atomics:
// fixed ascending-atom accumulation order). One block per token, thread per
// output channel; tok[] staged in LDS so the inner test is a scalar branch.
// ---------------------------------------------------------------------------
__global__ void k_segmean(const float* __restrict__ x, const int* __restrict__ tok,
                          const int* __restrict__ apt, float* __restrict__ ai) {
  __shared__ int stok[N_ATOM];
  int t = blockIdx.x;
  for (int a0 = threadIdx.x; a0 < N_ATOM; a0 += blockDim.x) stok[a0] = tok[a0];
  __syncthreads();
  int c = threadIdx.x;                    // blockDim = C_TOKEN
  float s = 0.f;
  for (int a0 = 0; a0 < N_ATOM; ++a0) {
    if (stok[a0] == t)                    // lane-invariant -> s_cbranch skip
      s += fmaxf(x[(size_t)a0 * C_TOKEN + c], 0.f);
  }
  ai[(size_t)t * C_TOKEN + c] = s / (float)apt[t];
}

// ---------------------------------------------------------------------------
static void gemm(const float* A, const float* W, const float* bias, float* C,
                 int M, int K, int N, hipStream_t s) {
  gemm_wmma<<<dim3((unsigned)(N / 16), (unsigned)(M / 64)), dim3(128), 0, s>>>(
      A, W, bias, C, M, K, N);
}

extern "C" void kernel_launch(void* const* d_in, const int* in_sizes, int n_in,
                              void* d_out, int out_size, void* d_ws, size_t ws_size,
                              hipStream_t stream) {
  (void)in_sizes; (void)n_in; (void)out_size; (void)ws_size;
  // Inputs: setup_inputs dict order; nested params dict flattened in jax
  // pytree (sorted-key) order.
  const float* ref_pos    = (const float*)d_in[0];
  const float* ref_mask   = (const float*)d_in[1];
  const float* ref_elem   = (const float*)d_in[2];
  const float* ref_charge = (const float*)d_in[3];
  const float* ref_chars  = (const float*)d_in[4];
  const float* ref_uid    = (const float*)d_in[5];
  const float* rl         = (const float*)d_in[6];
  const float* si_trunk   = (const float*)d_in[7];
  const float* zij        = (const float*)d_in[8];
  const float* W_feats = (const float*)d_in[9];
  const float* W_inv   = (const float*)d_in[10];
  const float* W_l     = (const float*)d_in[11];
  const float* W_m     = (const float*)d_in[12];
  const float* W_mlp1  = (const float*)d_in[13];
  const float* W_mlp2  = (const float*)d_in[14];
  const float* W_mlp3  = (const float*)d_in[15];
  const float* W_off   = (const float*)d_in[16];
  const float* W_qout  = (const float*)d_in[17];
  const float* W_r     = (const float*)d_in[18];
  const float* W_s     = (const float*)d_in[19];
  const float* W_vm    = (const float*)d_in[20];
  const float* W_z     = (const float*)d_in[21];
  const float* B_Wb    = (const float*)d_in[22];
  const float* B_Wg    = (const float*)d_in[23];
  const float* B_Wk    = (const float*)d_in[24];
  const float* B_Wo    = (const float*)d_in[25];
  const float* B_Wos   = (const float*)d_in[26];
  const float* B_Wq    = (const float*)d_in[27];
  const float* B_Wt1   = (const float*)d_in[28];
  const float* B_Wt2   = (const float*)d_in[29];
  const float* B_Wto   = (const float*)d_in[30];
  const float* B_Wts   = (const float*)d_in[31];
  const float* B_Wv    = (const float*)d_in[32];
  const float* B_a1_Wg = (const float*)d_in[33];
  const float* B_a1_Ws = (const float*)d_in[34];
  const float* B_a1_b  = (const float*)d_in[35];
  const float* B_a1_bg = (const float*)d_in[36];
  const float* B_a1_g  = (const float*)d_in[37];
  const float* B_a2_Wg = (const float*)d_in[38];
  const float* B_a2_Ws = (const float*)d_in[39];
  const float* B_a2_b  = (const float*)d_in[40];
  const float* B_a2_bg = (const float*)d_in[41];
  const float* B_a2_g  = (const float*)d_in[42];
  const float* B_bos   = (const float*)d_in[43];
  const float* B_bq    = (const float*)d_in[44];
  const float* B_bts   = (const float*)d_in[45];
  const float* B_lnp_b = (const float*)d_in[46];
  const float* B_lnp_g = (const float*)d_in[47];
  const float* lns_b   = (const float*)d_in[48];
  const float* lns_g   = (const float*)d_in[49];
  const float* lnz_b   = (const float*)d_in[50];
  const float* lnz_g   = (const float*)d_in[51];
  const int*   tok     = (const int*)d_in[52];
  const int*   apt     = (const int*)d_in[53];

  float* out = (float*)d_out;
  float* out_ai  = out;                                  // [256,384]
  float* out_ql  = out_ai + (size_t)N_TOKEN * C_TOKEN;   // [2048,128]
  float* out_cl  = out_ql + (size_t)N_ATOM * C_ATOM;     // [2048,128]
  float* out_plm = out_cl + (size_t)N_ATOM * C_ATOM;     // [2048,2048,16]

  float* ws = (float*)d_ws;
  size_t off = 0;
  auto alloc = [&](size_t n) { float* p = ws + off; off += n; return p; };
  float* SNTOK = alloc((size_t)N_TOKEN * C_TOKEN);
  float* STOK  = alloc((size_t)N_TOKEN * C_ATOM);
  float* Z     = alloc((size_t)N_TOKEN * N_TOKEN * C_PAIR);
  float* RC    = alloc((size_t)N_ATOM * C_ATOM);
  float* CWL   = alloc((size_t)N_ATOM * C_PAIR);
  float* CWM   = alloc((size_t)N_ATOM * C_PAIR);
  float* A     = alloc((size_t)N_ATOM * C_ATOM);
  float* LN0A  = alloc((size_t)N_ATOM * C_ATOM);
  float* SN    = alloc((size_t)N_ATOM * C_ATOM);
  float* T1    = alloc((size_t)N_ATOM * C_ATOM);
  float* T2    = alloc((size_t)N_ATOM * C_ATOM);
  float* AN    = alloc((size_t)N_ATOM * C_ATOM);
  float* QB    = alloc((size_t)N_ATOM * C_ATOM);
  float* KB    = alloc((size_t)N_ATOM * C_ATOM);
  float* VB    = alloc((size_t)N_ATOM * C_ATOM);
  float* GB    = alloc((size_t)N_ATOM * C_ATOM);
  float* SIGA  = alloc((size_t)N_ATOM * C_ATOM);
  float* SIGB  = alloc((size_t)N_ATOM * C_ATOM);
  float* OG    = alloc((size_t)N_ATOM * C_ATOM);
  float* OP    = alloc((size_t)N_ATOM * C_ATOM);
  float* BATT  = alloc((size_t)N_ATOM * C_ATOM);
  float* TT1   = alloc((size_t)N_ATOM * 2 * C_ATOM);
  float* TT2   = alloc((size_t)N_ATOM * 2 * C_ATOM);
  float* HTO   = alloc((size_t)N_ATOM * C_ATOM);
  float* XOUT  = alloc((size_t)N_ATOM * C_TOKEN);

  int NC = N_ATOM * C_ATOM;
  dim3 ew((NC + 255) / 256), tb(256);

  // ---- precompute ----
  k_ln<<<dim3(N_TOKEN / 4), dim3(128), 0, stream>>>(
      si_trunk, lns_g, lns_b, SNTOK, N_TOKEN, C_TOKEN);
  gemm(SNTOK, W_s, nullptr, STOK, N_TOKEN, C_TOKEN, C_ATOM, stream);
  k_embed<<<dim3(N_ATOM), dim3(C_ATOM), 0, stream>>>(
      ref_pos, ref_mask, ref_elem, ref_charge, ref_chars, ref_uid, rl,
      W_feats, W_r, STOK, tok, out_ql, out_cl, A);
  k_zproj<<<dim3(N_TOKEN), dim3(N_TOKEN), 0, stream>>>(zij, lnz_g, lnz_b, W_z, Z);
  k_relu<<<ew, tb, 0, stream>>>(out_cl, RC, NC);
  gemm(RC, W_l, nullptr, CWL, N_ATOM, C_ATOM, C_PAIR, stream);
  gemm(RC, W_m, nullptr, CWM, N_ATOM, C_ATOM, C_PAIR, stream);
  k_plm<<<dim3(N_ATOM / 128, N_ATOM), dim3(256), 0, stream>>>(
      ref_pos, ref_uid, tok, W_off, W_inv, W_vm, Z, CWL, CWM,
      W_mlp1, W_mlp2, W_mlp3, out_plm);

  // ---- transformer blocks ----
  for (int b = 0; b < 3; ++b) {
    const float* a1_g  = B_a1_g  + b * C_ATOM;
    const float* a1_b  = B_a1_b  + b * C_ATOM;
    const float* a1_Wg = B_a1_Wg + b * C_ATOM * C_ATOM;
    const float* a1_bg = B_a1_bg + b * C_ATOM;
    const float* a1_Ws = B_a1_Ws + b * C_ATOM * C_ATOM;
    const float* Wq    = B_Wq    + b * C_ATOM * C_ATOM;
    const float* bq    = B_bq    + b * C_ATOM;
    const float* Wk    = B_Wk    + b * C_ATOM * C_ATOM;
    const float* Wv    = B_Wv    + b * C_ATOM * C_ATOM;
    const float* Wg    = B_Wg    + b * C_ATOM * C_ATOM;
    const float* Wo    = B_Wo    + b * C_ATOM * C_ATOM;
    const float* Wos   = B_Wos   + b * C_ATOM * C_ATOM;
    const float* bos   = B_bos   + b * C_ATOM;
    const float* lnp_g = B_lnp_g + b * C_PAIR;
    const float* lnp_b = B_lnp_b + b * C_PAIR;
    const float* Wb    = B_Wb    + b * C_PAIR * NHEADS;
    const float* a2_g  = B_a2_g  + b * C_ATOM;
    const float* a2_b  = B_a2_b  + b * C_ATOM;
    const float* a2_Wg = B_a2_Wg + b * C_ATOM * C_ATOM;
    const float* a2_bg = B_a2_bg + b * C_ATOM;
    const float* a2_Ws = B_a2_Ws + b * C_ATOM * C_ATOM;
    const float* Wt1   = B_Wt1   + b * C_ATOM * 2 * C_ATOM;
    const float* Wt2   = B_Wt2   + b * C_ATOM * 2 * C_ATOM;
    const float* Wto   = B_Wto   + b * 2 * C_ATOM * C_ATOM;
    const float* Wts   = B_Wts   + b * C_ATOM * C_ATOM;
    const float* bts   = B_bts   + b * C_ATOM;

    // adaLN #1 -> an
    k_ln<<<dim3(N_ATOM / 4), dim3(128), 0, stream>>>(out_cl, a1_g, a1_b, SN, N_ATOM, C_ATOM);
    gemm(SN, a1_Wg, nullptr, T1, N_ATOM, C_ATOM, C_ATOM, stream);
    gemm(SN, a1_Ws, nullptr, T2, N_ATOM, C_ATOM, C_ATOM, stream);
    k_ln<<<dim3(N_ATOM / 4), dim3(128), 0, stream>>>(A, nullptr, nullptr, LN0A, N_ATOM, C_ATOM);
    k_adaln<<<ew, tb, 0, stream>>>(T1, a1_bg, LN0A, T2, AN, NC, C_ATOM);
    // q,k,v,gate
    gemm(AN, Wq, bq, QB, N_ATOM, C_ATOM, C_ATOM, stream);
    gemm(AN, Wk, nullptr, KB, N_ATOM, C_ATOM, C_ATOM, stream);
    gemm(AN, Wv, nullptr, VB, N_ATOM, C_ATOM, C_ATOM, stream);
    gemm(AN, Wg, nullptr, GB, N_ATOM, C_ATOM, C_ATOM, stream);
    k_sigmoid<<<ew, tb, 0, stream>>>(GB, NC);
    gemm(out_cl, Wos, bos, SIGA, N_ATOM, C_ATOM, C_ATOM, stream);
    k_sigmoid<<<ew, tb, 0, stream>>>(SIGA, NC);
    // local attention + output proj
    k_attn<<<dim3(N_ATOM / 32, NHEADS / 2), dim3(64), 0, stream>>>(
        QB, KB, VB, GB, out_plm, lnp_g, lnp_b, Wb, OG);
    gemm(OG, Wo, nullptr, OP, N_ATOM, C_ATOM, C_ATOM, stream);
    k_mul<<<ew, tb, 0, stream>>>(SIGA, OP, BATT, NC);
    // adaLN #2 -> an2 (ln0(a) unchanged, reuse LN0A)
    k_ln<<<dim3(N_ATOM / 4), dim3(128), 0, stream>>>(out_cl, a2_g, a2_b, SN, N_ATOM, C_ATOM);
    gemm(SN, a2_Wg, nullptr, T1, N_ATOM, C_ATOM, C_ATOM, stream);
    gemm(SN, a2_Ws, nullptr, T2, N_ATOM, C_ATOM, C_ATOM, stream);
    k_adaln<<<ew, tb, 0, stream>>>(T1, a2_bg, LN0A, T2, AN, NC, C_ATOM);
    // conditioned transition
    gemm(AN, Wt1, nullptr, TT1, N_ATOM, C_ATOM, 2 * C_ATOM, stream);
    gemm(AN, Wt2, nullptr, TT2, N_ATOM, C_ATOM, 2 * C_ATOM, stream);
    k_silu_mul<<<dim3((2 * NC + 255) / 256), tb, 0, stream>>>(TT1, TT2, TT1, 2 * NC);
    gemm(TT1, Wto, nullptr, HTO, N_ATOM, 2 * C_ATOM, C_ATOM, stream);
    gemm(out_cl, Wts, bts, SIGB, N_ATOM, C_ATOM, C_ATOM, stream);
    k_sigmoid<<<ew, tb, 0, stream>>>(SIGB, NC);
    k_fma3<<<ew, tb, 0, stream>>>(BATT, SIGB, HTO, A, NC);   // a = b_attn + t
  }

  // ---- token aggregation ----
  gemm(A, W_qout, nullptr, XOUT, N_ATOM, C_ATOM, C_TOKEN, stream);
  k_segmean<<<dim3(N_TOKEN), dim3(C_TOKEN), 0, stream>>>(XOUT, tok, apt, out_ai);
}